// BoundingBoxDiscipline_12103217840697
// MI455X (gfx1250) — compile-verified
//
#include <hip/hip_runtime.h>

#define TB 16
#define TH_ 256
#define TW_ 256
#define TC 128
#define NPIX (TH_ * TW_)

#if defined(__HIP_DEVICE_COMPILE__) && \
    __has_builtin(__builtin_amdgcn_global_load_async_to_lds_b128) && \
    __has_builtin(__builtin_amdgcn_s_wait_asynccnt)
#define USE_ASYNC 1
#else
#define USE_ASYNC 0
#endif

#if USE_ASYNC
typedef int v4i __attribute__((ext_vector_type(4)));
typedef __attribute__((address_space(1))) v4i* gv4i_p;   // global (AS1) int4*
typedef __attribute__((address_space(3))) v4i* lv4i_p;   // LDS (AS3) int4*
#endif

__device__ __forceinline__ bool wave_any(int p) {
#if defined(__HIP_DEVICE_COMPILE__) && __has_builtin(__builtin_amdgcn_ballot_w32)
  return __builtin_amdgcn_ballot_w32(p) != 0u;
#else
  return __any(p);
#endif
}

// d_ws layout per batch (8 ints): [pymin, pxmin, pymax, pxmax, tymin, txmin, tymax, txmax]
__global__ void bbox_init(int* __restrict__ bb) {
  int i = blockIdx.x * blockDim.x + threadIdx.x;
  if (i < TB * 8) {
    int k = i & 7;
    bb[i] = (k == 0 || k == 1 || k == 4 || k == 5) ? 0x7fffffff : -1;
  }
}

__global__ __launch_bounds__(256) void bbox_scan(const float4* __restrict__ pred,
                                                 const float4* __restrict__ tru,
                                                 int* __restrict__ bb) {
  const int b    = blockIdx.y;
  const int lane = threadIdx.x & 31;
  const int wave = threadIdx.x >> 5;
  const int wpb  = gridDim.x * (blockDim.x >> 5);           // waves per batch
  const int gw   = blockIdx.x * (blockDim.x >> 5) + wave;   // this wave's id in batch

  int pymin = 0x7fffffff, pxmin = 0x7fffffff, pymax = -1, pxmax = -1;
  int tymin = 0x7fffffff, txmin = 0x7fffffff, tymax = -1, txmax = -1;

  const size_t base4 = (size_t)b * NPIX * (TC / 4);  // batch base, float4 units

#if USE_ASYNC
  // 8 waves x 2 stages x 2 tensors x 32 float4 (512B) = 16KB LDS
  __shared__ float4 sb[8 * 2 * 2 * (TC / 4)];
  int stage = 0;
  if (gw < NPIX) {
    // prologue: stage 0 of first pixel
    {
      const float4* gp = pred + base4 + (size_t)gw * (TC / 4) + lane;
      const float4* gt = tru  + base4 + (size_t)gw * (TC / 4) + lane;
      __builtin_amdgcn_global_load_async_to_lds_b128(
          (gv4i_p)const_cast<float4*>(gp),
          (lv4i_p)&sb[((wave * 2 + 0) * 2 + 0) * 32 + lane], 0, 0);
      __builtin_amdgcn_global_load_async_to_lds_b128(
          (gv4i_p)const_cast<float4*>(gt),
          (lv4i_p)&sb[((wave * 2 + 0) * 2 + 1) * 32 + lane], 0, 0);
    }
    for (int pix = gw; pix < NPIX; pix += wpb) {
      int np = pix + wpb;
      if (np >= NPIX) np = pix;  // dummy re-issue into the other stage on last iter
      const int ns = stage ^ 1;
      const float4* gp = pred + base4 + (size_t)np * (TC / 4) + lane;
      const float4* gt = tru  + base4 + (size_t)np * (TC / 4) + lane;
      __builtin_amdgcn_global_load_async_to_lds_b128(
          (gv4i_p)const_cast<float4*>(gp),
          (lv4i_p)&sb[((wave * 2 + ns) * 2 + 0) * 32 + lane], 0, 0);
      __builtin_amdgcn_global_load_async_to_lds_b128(
          (gv4i_p)const_cast<float4*>(gt),
          (lv4i_p)&sb[((wave * 2 + ns) * 2 + 1) * 32 + lane], 0, 0);
      // <=2 outstanding: the pair just issued. Current stage's pair is complete
      // (async loads of one wave complete in order).
      __builtin_amdgcn_s_wait_asynccnt(2);
      __asm__ __volatile__("" ::: "memory");
      const float4 vp = sb[((wave * 2 + stage) * 2 + 0) * 32 + lane];
      const float4 vt = sb[((wave * 2 + stage) * 2 + 1) * 32 + lane];
      const float mp = fmaxf(fmaxf(vp.x, vp.y), fmaxf(vp.z, vp.w));
      const float mt = fmaxf(fmaxf(vt.x, vt.y), fmaxf(vt.z, vt.w));
      const int y = pix >> 8, x = pix & (TW_ - 1);
      if (wave_any(mp > 0.3f)) {
        if (y < pymin) pymin = y; if (y > pymax) pymax = y;
        if (x < pxmin) pxmin = x; if (x > pxmax) pxmax = x;
      }
      if (wave_any(mt > 0.5f)) {
        if (y < tymin) tymin = y; if (y > tymax) tymax = y;
        if (x < txmin) txmin = x; if (x > txmax) txmax = x;
      }
      stage = ns;
    }
    __builtin_amdgcn_s_wait_asynccnt(0);
  }
#else
  for (int pix = gw; pix < NPIX; pix += wpb) {
    const float4 vp = pred[base4 + (size_t)pix * (TC / 4) + lane];
    const float4 vt = tru [base4 + (size_t)pix * (TC / 4) + lane];
    const float mp = fmaxf(fmaxf(vp.x, vp.y), fmaxf(vp.z, vp.w));
    const float mt = fmaxf(fmaxf(vt.x, vt.y), fmaxf(vt.z, vt.w));
    const int y = pix >> 8, x = pix & (TW_ - 1);
    if (wave_any(mp > 0.3f)) {
      if (y < pymin) pymin = y; if (y > pymax) pymax = y;
      if (x < pxmin) pxmin = x; if (x > pxmax) pxmax = x;
    }
    if (wave_any(mt > 0.5f)) {
      if (y < tymin) tymin = y; if (y > tymax) tymax = y;
      if (x < txmin) txmin = x; if (x > txmax) txmax = x;
    }
  }
#endif

  // bbox values are wave-uniform (ballot result is uniform); lane 0 commits.
  if (lane == 0) {
    int* q = bb + b * 8;
    if (pymax >= 0) {
      atomicMin(&q[0], pymin); atomicMin(&q[1], pxmin);
      atomicMax(&q[2], pymax); atomicMax(&q[3], pxmax);
    }
    if (tymax >= 0) {
      atomicMin(&q[4], tymin); atomicMin(&q[5], txmin);
      atomicMax(&q[6], tymax); atomicMax(&q[7], txmax);
    }
  }
}

__global__ void bbox_final(const int* __restrict__ bb, float* __restrict__ out) {
  if (threadIdx.x == 0 && blockIdx.x == 0) {
    float s = 0.f;
    for (int b = 0; b < TB; ++b) {
      const int* q = bb + b * 8;
      float py0, px0, py1, px1, ty0, tx0, ty1, tx1;
      if (q[2] < 0) { py0 = 0.f; px0 = 0.f; py1 = 1.f; px1 = 1.f; }
      else { py0 = (float)q[0]; px0 = (float)q[1]; py1 = (float)q[2]; px1 = (float)q[3]; }
      if (q[6] < 0) { ty0 = 0.f; tx0 = 0.f; ty1 = 1.f; tx1 = 1.f; }
      else { ty0 = (float)q[4]; tx0 = (float)q[5]; ty1 = (float)q[6]; tx1 = (float)q[7]; }
      const float pa = (py1 - py0 + 1.f) * (px1 - px0 + 1.f);
      const float ta = (ty1 - ty0 + 1.f) * (tx1 - tx0 + 1.f);
      const float area_pen = fmaxf(pa - ta, 0.f) / (ta + 1.f);
      const float dy = (py0 + py1) * 0.5f - (ty0 + ty1) * 0.5f;
      const float dx = (px0 + px1) * 0.5f - (tx0 + tx1) * 0.5f;
      const float center = sqrtf(dy * dy + dx * dx) * (1.0f / 20.0f);
      s += area_pen + center;
    }
    out[0] = 0.05f * (s * (1.0f / TB));
  }
}

extern "C" void kernel_launch(void* const* d_in, const int* in_sizes, int n_in,
                              void* d_out, int out_size, void* d_ws, size_t ws_size,
                              hipStream_t stream) {
  const float4* pred = (const float4*)d_in[0];
  const float4* tru  = (const float4*)d_in[1];
  int*   bb  = (int*)d_ws;
  float* out = (float*)d_out;

  bbox_init<<<dim3(1), dim3(128), 0, stream>>>(bb);
  bbox_scan<<<dim3(64, TB, 1), dim3(256), 0, stream>>>(pred, tru, bb);
  bbox_final<<<dim3(1), dim3(32), 0, stream>>>(bb, out);
}